// e3nn_76441827934642
// MI455X (gfx1250) — compile-verified
//
#include <hip/hip_runtime.h>
#include <hip/hip_bf16.h>

// e3nn GNN layer for MI455X (gfx1250, wave32, WMMA).
// fp32 everywhere (reference is fp32); dense GEMMs use V_WMMA_F32_16X16X4_F32.

#define NNODE 50000
#define NEDGE 600000
#define IN_F  92
#define MUL   32
#define NB    10
#define RH    100
#define WIN   400
#define FEATW 288   // 32 + 96 + 160

typedef float v2f __attribute__((ext_vector_type(2)));
typedef float v8f __attribute__((ext_vector_type(8)));

// ---------------- zero workspace accumulator ----------------
__global__ void zero_kernel(float* __restrict__ p, int n) {
    int i = blockIdx.x * blockDim.x + threadIdx.x;
    if (i < n) p[i] = 0.0f;
}

// ---------------- node embedding: x = node_attrs @ embed_W ----------------
// one wave per 16-node tile; 2 N-tiles (32 cols), K=92 (23 steps of 4)
__global__ __launch_bounds__(256) void embed_kernel(const float* __restrict__ na,
                                                    const float* __restrict__ W,
                                                    float* __restrict__ x) {
    int wave = (blockIdx.x * blockDim.x + threadIdx.x) >> 5;
    int lane = threadIdx.x & 31;
    if (wave >= NNODE / 16) return;
    int m0 = wave * 16;
    int rowA = m0 + (lane & 15);
    int kOff = (lane >> 4) * 2;     // A/B fragment K selector (lanes 16-31 hold K+2)
    int colLane = lane & 15;
    v8f acc0 = {}; v8f acc1 = {};
    for (int kb = 0; kb < IN_F; kb += 4) {
        v2f a, b0, b1;
        a.x = na[rowA * IN_F + kb + kOff];
        a.y = na[rowA * IN_F + kb + kOff + 1];
        b0.x = W[(kb + kOff)     * MUL + colLane];
        b0.y = W[(kb + kOff + 1) * MUL + colLane];
        b1.x = W[(kb + kOff)     * MUL + 16 + colLane];
        b1.y = W[(kb + kOff + 1) * MUL + 16 + colLane];
        acc0 = __builtin_amdgcn_wmma_f32_16x16x4_f32(false, a, false, b0, (short)0, acc0, false, false);
        acc1 = __builtin_amdgcn_wmma_f32_16x16x4_f32(false, a, false, b1, (short)0, acc1, false, false);
    }
    int rBase = m0 + (lane >> 4) * 8;   // C layout: row = v + 8*lane[4]
    for (int v = 0; v < 8; ++v) {
        x[(rBase + v) * MUL + colLane]      = acc0[v];
        x[(rBase + v) * MUL + 16 + colLane] = acc1[v];
    }
}

// ---------------- per-edge: radial MLP + SH messages + scatter ----------------
// one wave per edge; MUL==32 so lane == mul channel.
__global__ __launch_bounds__(256) void edge_kernel(const float* __restrict__ pos,
                                                   const float* __restrict__ shifts,
                                                   const int*   __restrict__ ei,
                                                   const float* __restrict__ x,
                                                   const float* __restrict__ Wr1,
                                                   const float* __restrict__ Wr2,
                                                   float* __restrict__ feat) {
    int wave = (blockIdx.x * blockDim.x + threadIdx.x) >> 5;
    int lane = threadIdx.x & 31;
    if (wave >= NEDGE) return;
    int e = wave;
    int src = ei[e];
    int dst = ei[NEDGE + e];

    float vx = pos[dst * 3 + 0] - pos[src * 3 + 0] + shifts[e * 3 + 0];
    float vy = pos[dst * 3 + 1] - pos[src * 3 + 1] + shifts[e * 3 + 1];
    float vz = pos[dst * 3 + 2] - pos[src * 3 + 2] + shifts[e * 3 + 2];
    float r = sqrtf(vx * vx + vy * vy + vz * vz);
    float inv = 1.0f / (r + 1e-9f);
    float ux = vx * inv, uy = vy * inv, uz = vz * inv;

    // Gaussian radial basis (all lanes redundantly; 10 values)
    const float sqrt10 = 3.1622776601683795f;
    float rb[NB];
#pragma unroll
    for (int i = 0; i < NB; ++i) {
        float c = 5.0f * (float)i / 9.0f;       // linspace(0, RMAX, NB)
        float t = (r - c) * 2.0f;               // * NB/RMAX
        rb[i] = expf(-t * t) * sqrt10;
    }

    // hidden = silu(rb @ Wr1): 100 cols, 4 per lane (slot 3 only lanes 0-3)
    float hl[4];
#pragma unroll
    for (int s = 0; s < 4; ++s) {
        int c = lane + 32 * s;
        float v = 0.0f;
        if (c < RH) {
            float z = 0.0f;
#pragma unroll
            for (int i = 0; i < NB; ++i) z += rb[i] * Wr1[i * RH + c];
            v = z / (1.0f + expf(-z));
        }
        hl[s] = v;
    }

    // w = hidden @ Wr2 (100 x 96); lane owns cols {lane, lane+32, lane+64}
    float w0 = 0.0f, w1 = 0.0f, w2 = 0.0f;
    for (int k = 0; k < RH; ++k) {
        float hk = __shfl(hl[k >> 5], k & 31, 32);
        w0 += hk * Wr2[k * (3 * MUL) + lane];
        w1 += hk * Wr2[k * (3 * MUL) + 32 + lane];
        w2 += hk * Wr2[k * (3 * MUL) + 64 + lane];
    }

    float xs = x[src * MUL + lane];
    float* fb = feat + (size_t)dst * FEATW;

    const float s3  = 1.7320508075688772f;
    const float s5  = 2.23606797749979f;
    const float s15 = 3.872983346207417f;

    // l=0
    atomicAdd(fb + lane, w0 * xs);
    // l=1: y1 = s3 * (ux, uy, uz); flat index 32 + mul*3 + c
    float b1 = w1 * xs;
    atomicAdd(fb + 32 + lane * 3 + 0, b1 * s3 * ux);
    atomicAdd(fb + 32 + lane * 3 + 1, b1 * s3 * uy);
    atomicAdd(fb + 32 + lane * 3 + 2, b1 * s3 * uz);
    // l=2: flat index 128 + mul*5 + c
    float g = w2 * xs;
    float y2c0 = s15 * ux * uy;
    float y2c1 = s15 * uy * uz;
    float y2c2 = 0.5f * s5 * (3.0f * uz * uz - 1.0f);
    float y2c3 = s15 * ux * uz;
    float y2c4 = 0.5f * s15 * (ux * ux - uy * uy);
    atomicAdd(fb + 128 + lane * 5 + 0, g * y2c0);
    atomicAdd(fb + 128 + lane * 5 + 1, g * y2c1);
    atomicAdd(fb + 128 + lane * 5 + 2, g * y2c2);
    atomicAdd(fb + 128 + lane * 5 + 3, g * y2c3);
    atomicAdd(fb + 128 + lane * 5 + 4, g * y2c4);
}

// ---------------- readout: dos = tanh(prelu(f@Wd1)@Wd2), scaling = relu(prelu(f@Ws1)@Ws2)
#define FSTR 292   // 288 padded (bank spread)
#define HSTR 516   // 512 padded
#define SSTR 116   // 112 padded
__global__ __launch_bounds__(256) void readout_kernel(const float* __restrict__ feat,
                                                      const float* __restrict__ Wd1,
                                                      const float* __restrict__ bd1,
                                                      const float* __restrict__ ad1,
                                                      const float* __restrict__ Wd2,
                                                      const float* __restrict__ bd2,
                                                      const float* __restrict__ Ws1,
                                                      const float* __restrict__ bs1,
                                                      const float* __restrict__ as1,
                                                      const float* __restrict__ Ws2,
                                                      const float* __restrict__ bs2,
                                                      float* __restrict__ out) {
    __shared__ float fA[16 * FSTR];
    __shared__ float hB[16 * HSTR];
    __shared__ float sB[16 * SSTR];

    int tile = blockIdx.x;
    int m0 = tile * 16;
    int tid = threadIdx.x;
    int wave = tid >> 5, lane = tid & 31;

    const float inv12 = 0.28867513459481287f;  // 1/sqrt(NUM_NEIGH)

    // stage 0: load feat tile (scaled)
    for (int i = tid; i < 16 * FEATW; i += 256) {
        int rr = i / FEATW, cc = i % FEATW;
        fA[rr * FSTR + cc] = feat[(size_t)(m0 + rr) * FEATW + cc] * inv12;
    }
    __syncthreads();

    int rowA = lane & 15;
    int kOff = (lane >> 4) * 2;
    int colLane = lane & 15;
    int rBase = (lane >> 4) * 8;
    float a1 = ad1[0];

    // stage 1: h = prelu(fA @ Wd1 + bd1)  [16x500 -> padded 512, 32 N-tiles / 8 waves]
    for (int it = 0; it < 4; ++it) {
        int nt = wave + 8 * it;
        int col = nt * 16 + colLane;
        v8f acc = {};
        for (int kb = 0; kb < FEATW; kb += 4) {
            v2f a, b;
            a.x = fA[rowA * FSTR + kb + kOff];
            a.y = fA[rowA * FSTR + kb + kOff + 1];
            if (col < 500) {
                b.x = Wd1[(kb + kOff)     * 500 + col];
                b.y = Wd1[(kb + kOff + 1) * 500 + col];
            } else { b.x = 0.0f; b.y = 0.0f; }
            acc = __builtin_amdgcn_wmma_f32_16x16x4_f32(false, a, false, b, (short)0, acc, false, false);
        }
        for (int v = 0; v < 8; ++v) {
            float val = 0.0f;
            if (col < 500) {
                float z = acc[v] + bd1[col];
                val = z > 0.0f ? z : a1 * z;
            }
            hB[(rBase + v) * HSTR + col] = val;
        }
    }
    __syncthreads();

    // stage 2: dos = tanh(h @ Wd2 + bd2)  [K=500 exact (125 steps), 25 N-tiles]
    for (int it = 0; it < 4; ++it) {
        int nt = wave + 8 * it;
        if (nt >= 25) break;               // wave-uniform
        int col = nt * 16 + colLane;
        v8f acc = {};
        for (int kb = 0; kb < 500; kb += 4) {
            v2f a, b;
            a.x = hB[rowA * HSTR + kb + kOff];
            a.y = hB[rowA * HSTR + kb + kOff + 1];
            b.x = Wd2[(kb + kOff)     * WIN + col];
            b.y = Wd2[(kb + kOff + 1) * WIN + col];
            acc = __builtin_amdgcn_wmma_f32_16x16x4_f32(false, a, false, b, (short)0, acc, false, false);
        }
        for (int v = 0; v < 8; ++v) {
            out[(size_t)(m0 + rBase + v) * 401 + col] = tanhf(acc[v] + bd2[col]);
        }
    }

    // stage 3: s = prelu(fA @ Ws1 + bs1)  [16x100 -> padded 112, waves 0..6]
    if (wave < 7) {
        int col = wave * 16 + colLane;
        v8f acc = {};
        for (int kb = 0; kb < FEATW; kb += 4) {
            v2f a, b;
            a.x = fA[rowA * FSTR + kb + kOff];
            a.y = fA[rowA * FSTR + kb + kOff + 1];
            if (col < 100) {
                b.x = Ws1[(kb + kOff)     * 100 + col];
                b.y = Ws1[(kb + kOff + 1) * 100 + col];
            } else { b.x = 0.0f; b.y = 0.0f; }
            acc = __builtin_amdgcn_wmma_f32_16x16x4_f32(false, a, false, b, (short)0, acc, false, false);
        }
        float a2 = as1[0];
        for (int v = 0; v < 8; ++v) {
            float val = 0.0f;
            if (col < 100) {
                float z = acc[v] + bs1[col];
                val = z > 0.0f ? z : a2 * z;
            }
            sB[(rBase + v) * SSTR + col] = val;
        }
    }
    __syncthreads();

    // stage 4: scaling = relu(s @ Ws2 + bs2)  [100 -> 1, 16 rows]
    if (tid < 16) {
        float sum = bs2[0];
        for (int k = 0; k < 100; ++k) sum += sB[tid * SSTR + k] * Ws2[k];
        out[(size_t)(m0 + tid) * 401 + 400] = sum > 0.0f ? sum : 0.0f;
    }
}

extern "C" void kernel_launch(void* const* d_in, const int* in_sizes, int n_in,
                              void* d_out, int out_size, void* d_ws, size_t ws_size,
                              hipStream_t stream) {
    const float* node_attrs = (const float*)d_in[0];
    const float* positions  = (const float*)d_in[1];
    const float* shifts     = (const float*)d_in[2];
    const float* embed_W    = (const float*)d_in[3];
    const float* W_rad1     = (const float*)d_in[4];
    const float* W_rad2     = (const float*)d_in[5];
    const float* Wd1        = (const float*)d_in[6];
    const float* bd1        = (const float*)d_in[7];
    const float* ad1        = (const float*)d_in[8];
    const float* Wd2        = (const float*)d_in[9];
    const float* bd2        = (const float*)d_in[10];
    const float* Ws1        = (const float*)d_in[11];
    const float* bs1        = (const float*)d_in[12];
    const float* as1        = (const float*)d_in[13];
    const float* Ws2        = (const float*)d_in[14];
    const float* bs2        = (const float*)d_in[15];
    const int*   edge_index = (const int*)d_in[16];
    float* out = (float*)d_out;

    // workspace: feat (N*288) then x (N*32)
    float* feat = (float*)d_ws;
    float* x    = feat + (size_t)NNODE * FEATW;

    int nFeat = NNODE * FEATW;
    zero_kernel<<<(nFeat + 255) / 256, 256, 0, stream>>>(feat, nFeat);

    int embedWaves = NNODE / 16;                 // 3125
    embed_kernel<<<(embedWaves + 7) / 8, 256, 0, stream>>>(node_attrs, embed_W, x);

    edge_kernel<<<NEDGE / 8, 256, 0, stream>>>(positions, shifts, edge_index,
                                               x, W_rad1, W_rad2, feat);

    readout_kernel<<<NNODE / 16, 256, 0, stream>>>(feat, Wd1, bd1, ad1, Wd2, bd2,
                                                   Ws1, bs1, as1, Ws2, bs2, out);
}